// CaseIDtoFeatureArchCore_71124658422108
// MI455X (gfx1250) — compile-verified
//
#include <hip/hip_runtime.h>
#include <hip/hip_bf16.h>
#include <stdint.h>

// out[i, :] = feature_array[round(x[i]), :]  (one-hot matmul == gather)
// Memory-bound: ~52 MB total traffic -> ~2.3 us floor at 23.3 TB/s HBM.
// CDNA5 path used: async global->LDS DMA (ASYNCcnt) to stage the 128 KB
// feature table into LDS once per block, then bank-conflict-free ds_load_b128
// gathers + coalesced b128 global stores.

#define D_FEAT 64  // feature dim (reference constant)

__global__ __launch_bounds__(256) void
caseid_gather_kernel(const float* __restrict__ x,
                     const float* __restrict__ table,
                     float* __restrict__ out,
                     int n, int total_case) {
    extern __shared__ float tbl[];  // total_case * D_FEAT floats (128 KB for 512x64)

    const int tid = threadIdx.x;

    // ---- Stage feature table into LDS via CDNA5 async global->LDS copies ----
    {
        const int nvec = (total_case * D_FEAT) >> 2;  // number of float4 chunks
        // Low 32 bits of the flat address of an LDS object are its LDS byte
        // offset (LDS aperture: addr[63:32]=shared_base, addr[31:0]=offset).
        const unsigned lds_base = (unsigned)(uintptr_t)tbl;
        for (int i = tid; i < nvec; i += (int)blockDim.x) {
            const unsigned lds_addr = lds_base + (unsigned)i * 16u;
            const unsigned long long gaddr =
                (unsigned long long)(uintptr_t)(table + (size_t)i * 4);
            // VDST = LDS byte address, VADDR = 64-bit global address (GV mode).
            asm volatile("global_load_async_to_lds_b128 %0, %1, off"
                         :: "v"(lds_addr), "v"(gaddr)
                         : "memory");
        }
        // Wait for this wave's async DMA to land in LDS, then sync the block.
#if __has_builtin(__builtin_amdgcn_s_wait_asynccnt)
        __builtin_amdgcn_s_wait_asynccnt(0);
#else
        asm volatile("s_wait_asynccnt 0x0" ::: "memory");
#endif
        __syncthreads();
    }

    // ---- Gather: 16 lanes per row, one float4 (16 B) per lane ----
    // Wave of 32 lanes: lanes 0-15 -> row A segs 0-15 (touch LDS banks 0..63
    // exactly once), lanes 16-31 -> row B likewise. ds_load_b128 is inherently
    // 2-phase for 32 lanes, so this is conflict-free. Stores: 512 contiguous
    // bytes per wave -> fully coalesced.
    const int seg = tid & 15;        // float4 segment within the row
    const int rlane = tid >> 4;      // row within this block's chunk of 16
    const int rowsPerIter = (int)blockDim.x >> 4;
    const int stride = (int)gridDim.x * rowsPerIter;

    for (int row = (int)blockIdx.x * rowsPerIter + rlane; row < n; row += stride) {
        const float xv = x[row];
        const int idx = (int)(xv + 0.5f);  // x is integer-valued, >= 0
        float4 v;
        const float d = xv - (float)idx;
        if (idx >= 0 && idx < total_case && d > -0.5f && d < 0.5f) {
            v = *(const float4*)(&tbl[(size_t)idx * D_FEAT + seg * 4]);
        } else {
            v = make_float4(0.f, 0.f, 0.f, 0.f);  // no matching case -> zero row
        }
        *(float4*)(&out[(size_t)row * D_FEAT + seg * 4]) = v;
    }
}

extern "C" void kernel_launch(void* const* d_in, const int* in_sizes, int n_in,
                              void* d_out, int out_size, void* d_ws, size_t ws_size,
                              hipStream_t stream) {
    (void)n_in; (void)d_ws; (void)ws_size; (void)out_size;

    const float* x     = (const float*)d_in[0];   // [N, 1] float32 (integer-valued)
    const float* table = (const float*)d_in[1];   // [total_case, 64] float32
    float*       out   = (float*)d_out;           // [N, 64] float32

    const int n          = in_sizes[0];           // N (N*1 flat)
    const int feat_elems = in_sizes[1];           // total_case * 64
    const int total_case = feat_elems / D_FEAT;

    // 256 threads = 8 waves; 16 rows per block-iteration; grid-stride loop.
    // 128 KB dynamic LDS -> 2 blocks resident per WGP (320 KB LDS).
    int blocks = (n + 15) / 16;
    if (blocks > 512) blocks = 512;
    if (blocks < 1) blocks = 1;
    const size_t shmem = (size_t)feat_elems * sizeof(float);

    caseid_gather_kernel<<<blocks, 256, shmem, stream>>>(x, table, out, n, total_case);
}